// StackedConv_24592982737045
// MI455X (gfx1250) — compile-verified
//
#include <hip/hip_runtime.h>
#include <hip/hip_bf16.h>

typedef float v2f __attribute__((ext_vector_type(2)));
typedef float v8f __attribute__((ext_vector_type(8)));

#define DFEAT 128
#define SCAN_CHUNK 1024   // elements per scan block (256 thr x 4)

// ---------------------------------------------------------------------------
// Zero an int array (degree histogram).
// ---------------------------------------------------------------------------
__global__ void zero_int_kernel(int* __restrict__ p, int n) {
    int i = blockIdx.x * blockDim.x + threadIdx.x;
    if (i < n) p[i] = 0;
}

// ---------------------------------------------------------------------------
// CSR build step 1: degree histogram of dst (int atomics, 1 per edge).
// ---------------------------------------------------------------------------
__global__ void count_kernel(const int* __restrict__ dst,
                             int* __restrict__ cnt, int nedges) {
    int e = blockIdx.x * blockDim.x + threadIdx.x;
    if (e < nedges) atomicAdd(&cnt[dst[e]], 1);
}

// ---------------------------------------------------------------------------
// CSR build step 2a: per-1024-chunk exclusive scan (256 thr x 4 elems),
// block totals to bsum[].
// ---------------------------------------------------------------------------
__global__ void __launch_bounds__(256)
scan_partial_kernel(const int* __restrict__ cnt, int* __restrict__ off,
                    int* __restrict__ bsum, int n) {
    __shared__ int ls[256];
    const int t = threadIdx.x;
    const int base = blockIdx.x * SCAN_CHUNK;
    int v[4]; int s = 0;
#pragma unroll
    for (int j = 0; j < 4; ++j) {
        int i = base + t * 4 + j;
        v[j] = (i < n) ? cnt[i] : 0;
        s += v[j];
    }
    ls[t] = s; __syncthreads();
    for (int d = 1; d < 256; d <<= 1) {          // Hillis-Steele inclusive
        int add = (t >= d) ? ls[t - d] : 0; __syncthreads();
        ls[t] += add; __syncthreads();
    }
    if (t == 255) bsum[blockIdx.x] = ls[255];
    int run = (t == 0) ? 0 : ls[t - 1];
#pragma unroll
    for (int j = 0; j < 4; ++j) {
        int i = base + t * 4 + j;
        if (i < n) off[i] = run;
        run += v[j];
    }
}

// ---------------------------------------------------------------------------
// CSR build step 2b: exclusive scan of block sums (nb <= 256, one block).
// ---------------------------------------------------------------------------
__global__ void __launch_bounds__(256)
scan_bsum_kernel(int* __restrict__ bsum, int nb) {
    __shared__ int ls[256];
    const int t = threadIdx.x;
    ls[t] = (t < nb) ? bsum[t] : 0; __syncthreads();
    for (int d = 1; d < 256; d <<= 1) {
        int add = (t >= d) ? ls[t - d] : 0; __syncthreads();
        ls[t] += add; __syncthreads();
    }
    if (t < nb) bsum[t] = (t == 0) ? 0 : ls[t - 1];
}

// ---------------------------------------------------------------------------
// CSR build step 2c: add scanned block offsets; off[n] = E sentinel.
// ---------------------------------------------------------------------------
__global__ void scan_add_kernel(int* __restrict__ off, const int* __restrict__ bsum,
                                int n, int total) {
    int i = blockIdx.x * blockDim.x + threadIdx.x;
    if (i < n) off[i] += bsum[i / SCAN_CHUNK];
    if (i == 0) off[n] = total;
}

// ---------------------------------------------------------------------------
// CSR build step 3: bucket-fill sorted_src. After this kernel off[n] holds the
// END offset of node n (start = off[n-1], start of node 0 = 0) -> no cursor.
// ---------------------------------------------------------------------------
__global__ void fill_kernel(const int* __restrict__ src, const int* __restrict__ dst,
                            int* __restrict__ off, int* __restrict__ ssrc,
                            int nedges) {
    int e = blockIdx.x * blockDim.x + threadIdx.x;
    if (e < nedges) {
        int pos = atomicAdd(&off[dst[e]], 1);
        ssrc[pos] = src[e];
    }
}

// ---------------------------------------------------------------------------
// Atomic-free aggregation: one wave32 per node. Lane owns a float4 chunk of
// the 128-float row; accumulate deg gathered rows (512B coalesced, L2-hits)
// in registers, single 512B store. Node/src indices are wave-uniform ->
// readfirstlane forces the scalar (s_load) path for all index traffic.
// Every row is fully written, so no agg pre-zero is needed.
// ---------------------------------------------------------------------------
__global__ void __launch_bounds__(256)
gather_agg_kernel(const float* __restrict__ x, const int* __restrict__ ssrc,
                  const int* __restrict__ off, float* __restrict__ agg,
                  int nnodes) {
    int node = __builtin_amdgcn_readfirstlane(blockIdx.x * 8 + (threadIdx.x >> 5));
    if (node >= nnodes) return;                  // wave-uniform
    const int lane = threadIdx.x & 31;
    int beg = __builtin_amdgcn_readfirstlane(node ? off[node - 1] : 0);
    int end = __builtin_amdgcn_readfirstlane(off[node]);

    const float* xb = x + lane * 4;
    float4 acc = make_float4(0.f, 0.f, 0.f, 0.f);
    int e = beg;
    for (; e + 2 <= end; e += 2) {               // unroll x2 for MLP
        int s0 = __builtin_amdgcn_readfirstlane(ssrc[e]);
        int s1 = __builtin_amdgcn_readfirstlane(ssrc[e + 1]);
        const float4 v0 = *(const float4*)(xb + (size_t)s0 * DFEAT);
        const float4 v1 = *(const float4*)(xb + (size_t)s1 * DFEAT);
        acc.x += v0.x; acc.y += v0.y; acc.z += v0.z; acc.w += v0.w;
        acc.x += v1.x; acc.y += v1.y; acc.z += v1.z; acc.w += v1.w;
    }
    if (e < end) {
        int s0 = __builtin_amdgcn_readfirstlane(ssrc[e]);
        const float4 v0 = *(const float4*)(xb + (size_t)s0 * DFEAT);
        acc.x += v0.x; acc.y += v0.y; acc.z += v0.z; acc.w += v0.w;
    }
    *(float4*)(agg + (size_t)node * DFEAT + lane * 4) = acc;
}

// ---------------------------------------------------------------------------
// out[m,n] = relu( sum_k A[m,k] * W[n,k] + bias[n] )  via V_WMMA_F32_16X16X4_F32
// One wave = 16-row strip x 128 cols (8 v8f accumulators, 256 WMMAs).
// ---------------------------------------------------------------------------
__global__ void __launch_bounds__(256)
gemm_bias_relu_kernel(const float* __restrict__ A,
                      const float* __restrict__ W,
                      const float* __restrict__ bias,
                      float*       __restrict__ out,
                      int nrows) {
    const int wave  = threadIdx.x >> 5;
    const int lane  = threadIdx.x & 31;
    const int strip = blockIdx.x * 8 + wave;
    const int row0  = strip * 16;
    if (row0 + 16 > nrows) return;      // wave-uniform: EXEC stays all-1s

    const int half = lane >> 4;         // 0: K pair {0,1}; 1: K pair {2,3}
    const int lm   = lane & 15;         // M (for A) / N (for B)

    v8f c[8] = {};
    const float* arow = A + (size_t)(row0 + lm) * DFEAT;

    for (int k0 = 0; k0 < DFEAT; k0 += 4) {
        const int ka = k0 + half * 2;
        v2f a = *(const v2f*)(arow + ka);
#pragma unroll
        for (int t = 0; t < 8; ++t) {
            v2f bm = *(const v2f*)(W + (size_t)(t * 16 + lm) * DFEAT + ka);
            c[t] = __builtin_amdgcn_wmma_f32_16x16x4_f32(
                false, a, false, bm, (short)0, c[t], false, false);
        }
    }

#pragma unroll
    for (int t = 0; t < 8; ++t) {
        const int n  = t * 16 + lm;
        const float bv = bias[n];
#pragma unroll
        for (int v = 0; v < 8; ++v) {
            const int m = row0 + v + half * 8;
            float val = c[t][v] + bv;
            out[(size_t)m * DFEAT + n] = val > 0.f ? val : 0.f;
        }
    }
}

// ---------------------------------------------------------------------------
// Launch: build CSR once (shared by both layers), then
//   gather(x)->agg, wmma-gemm(W1)->d_out, gather(d_out)->agg, wmma-gemm(W2)->d_out
// All ws buffers are fully rewritten before being read each call (replay-safe).
// ---------------------------------------------------------------------------
extern "C" void kernel_launch(void* const* d_in, const int* in_sizes, int n_in,
                              void* d_out, int out_size, void* d_ws, size_t ws_size,
                              hipStream_t stream) {
    const float* x  = (const float*)d_in[0];
    const int*   ei = (const int*)  d_in[1];   // [2, E]: row0=src, row1=dst
    const float* W1 = (const float*)d_in[2];
    const float* b1 = (const float*)d_in[3];
    const float* W2 = (const float*)d_in[4];
    const float* b2 = (const float*)d_in[5];

    const int N = in_sizes[0] / DFEAT;   // 100000 (multiple of 16)
    const int E = in_sizes[1] / 2;       // 1600000
    const int* src = ei;
    const int* dst = ei + E;

    // Workspace layout
    float* agg  = (float*)d_ws;                       // N*128 f32   (51.2 MB)
    int*   cnt  = (int*)(agg + (size_t)N * DFEAT);    // N
    int*   off  = cnt + N;                            // N+1
    int*   bsum = off + N + 1;                        // up to 256
    int*   ssrc = bsum + 256;                         // E            (6.4 MB)

    float* out = (float*)d_out;

    const int nb = (N + SCAN_CHUNK - 1) / SCAN_CHUNK;   // 98 (<=256)
    dim3 blk(256);
    dim3 egrid((E + 255) / 256);
    dim3 ngrid((N + 255) / 256);
    dim3 agrid((N + 7) / 8);             // 8 waves/block, 1 node/wave
    dim3 ggrid((N / 16 + 7) / 8);

    // ---- CSR build (once, reused by both layers) ----
    zero_int_kernel<<<ngrid, blk, 0, stream>>>(cnt, N);
    count_kernel<<<egrid, blk, 0, stream>>>(dst, cnt, E);
    scan_partial_kernel<<<dim3(nb), blk, 0, stream>>>(cnt, off, bsum, N);
    scan_bsum_kernel<<<dim3(1), blk, 0, stream>>>(bsum, nb);
    scan_add_kernel<<<ngrid, blk, 0, stream>>>(off, bsum, N, E);
    fill_kernel<<<egrid, blk, 0, stream>>>(src, dst, off, ssrc, E);

    // ---- layer 1 ----
    gather_agg_kernel<<<agrid, blk, 0, stream>>>(x, ssrc, off, agg, N);
    gemm_bias_relu_kernel<<<ggrid, blk, 0, stream>>>(agg, W1, b1, out, N);

    // ---- layer 2 ----
    gather_agg_kernel<<<agrid, blk, 0, stream>>>(out, ssrc, off, agg, N);
    gemm_bias_relu_kernel<<<ggrid, blk, 0, stream>>>(agg, W2, b2, out, N);
}